// GaussianKernel_3410204033425
// MI455X (gfx1250) — compile-verified
//
#include <hip/hip_runtime.h>
#include <math.h>

// ---------------------------------------------------------------------------
// Gaussian kernel matrix on MI455X (gfx1250, wave32).
//   out[i,j] = exp(-d2[i,j] / (2*sigma2)),  d2 = ||xi||^2+||xj||^2-2*xi.xj
//   sigma2   = mean(d2) = (2/N)*sum_i ||xi||^2 - (2/N^2)*||sum_i xi||^2
// Main pass: V_WMMA_F32_16X16X4_F32 tiles of G = X*X^T fused with exp.
// ---------------------------------------------------------------------------

typedef float v2f __attribute__((ext_vector_type(2)));
typedef float v8f __attribute__((ext_vector_type(8)));

#define NPTS 2048
#define NF   128
#define ALPHA 1.0f

// workspace layout (in floats)
#define WS_SIGMA 0
#define WS_NORMS 16
#define WS_COLS  (16 + NPTS)
#define K1_BLOCKS 16   // 16 blocks * 128 col-partials = 2048 floats

// ---------------------------------------------------------------------------
// Kernel 1: per-row squared norms + per-block column-sum partials.
// 16 blocks x 256 threads (8 waves); each wave handles 16 rows; one row's 128
// floats = 32 lanes x float4 (fully coalesced b128 loads).
// ---------------------------------------------------------------------------
__global__ __launch_bounds__(256) void norms_cols_kernel(
    const float* __restrict__ X, float* __restrict__ norms,
    float* __restrict__ colpart) {
  __shared__ float cols[8 * NF];
  const int wave = threadIdx.x >> 5;
  const int lane = threadIdx.x & 31;
  const int rowsPerBlock = NPTS / K1_BLOCKS;     // 128
  const int rowsPerWave  = rowsPerBlock / 8;     // 16
  const int baseRow = blockIdx.x * rowsPerBlock + wave * rowsPerWave;

  float cx = 0.f, cy = 0.f, cz = 0.f, cw = 0.f;
  for (int r = 0; r < rowsPerWave; ++r) {
    const int row = baseRow + r;
    const float4 v = ((const float4*)(X + (size_t)row * NF))[lane];
    cx += v.x; cy += v.y; cz += v.z; cw += v.w;
    float s = v.x * v.x + v.y * v.y + v.z * v.z + v.w * v.w;
    // wave32 butterfly reduction (deterministic)
    for (int off = 16; off >= 1; off >>= 1) s += __shfl_xor(s, off, 32);
    if (lane == 0) norms[row] = s;
  }
  // each wave's column partial (cols lane*4 .. lane*4+3) -> LDS, fixed-order sum
  float* c = cols + wave * NF + lane * 4;
  c[0] = cx; c[1] = cy; c[2] = cz; c[3] = cw;
  __syncthreads();
  if (threadIdx.x < NF) {
    float s = 0.f;
    for (int w = 0; w < 8; ++w) s += cols[w * NF + threadIdx.x];
    colpart[blockIdx.x * NF + threadIdx.x] = s;
  }
}

// ---------------------------------------------------------------------------
// Kernel 2: sigma^2 = ALPHA * [ (2/N) sum(norms) - (2/N^2) ||colsum||^2 ]
// Single block, fixed-order tree reductions -> deterministic.
// ---------------------------------------------------------------------------
__global__ __launch_bounds__(256) void sigma_kernel(
    const float* __restrict__ norms, const float* __restrict__ colpart,
    float* __restrict__ sigma2) {
  __shared__ float red[256];
  const int tid = threadIdx.x;

  float s = 0.f;
  for (int i = tid; i < NPTS; i += 256) s += norms[i];
  red[tid] = s;
  __syncthreads();
  for (int off = 128; off >= 1; off >>= 1) {
    if (tid < off) red[tid] += red[tid + off];
    __syncthreads();
  }
  const float sumN = red[0];
  __syncthreads();

  float q = 0.f;
  if (tid < NF) {
    float cs = 0.f;
    for (int b = 0; b < K1_BLOCKS; ++b) cs += colpart[b * NF + tid];
    q = cs * cs;
  }
  red[tid] = q;
  __syncthreads();
  for (int off = 128; off >= 1; off >>= 1) {
    if (tid < off) red[tid] += red[tid + off];
    __syncthreads();
  }
  if (tid == 0) {
    const float invN = 1.0f / (float)NPTS;
    const float meanD2 = 2.0f * sumN * invN - 2.0f * red[0] * invN * invN;
    sigma2[0] = ALPHA * meanD2;
  }
}

// ---------------------------------------------------------------------------
// Kernel 3: main pass. One 16x16 output tile per wave via 32x
// V_WMMA_F32_16X16X4_F32 over K=128, fused norm/exp epilogue.
//
// A 16x4 f32 fragment layout (ISA 7.12.2): lanes 0-15 = rows M=0..15 holding
// K=kb+0 (v0), kb+1 (v1); lanes 16-31 = same rows holding K=kb+2, kb+3.
// => per-lane aligned float2 load at X[(r0+l16)*128 + kb + half*2].
// B = X^T column tile loads identically with base c0 (Gram symmetry).
// C/D: VGPR v holds row M = v + half*8, col N = c0 + l16.
// ---------------------------------------------------------------------------
__global__ __launch_bounds__(256) void gauss_wmma_kernel(
    const float* __restrict__ X, const float* __restrict__ norms,
    const float* __restrict__ sigma2, float* __restrict__ out) {
  const int wave = threadIdx.x >> 5;
  const int lane = threadIdx.x & 31;
  const int half = lane >> 4;
  const int l16  = lane & 15;

  const int tile = blockIdx.x * 8 + wave;     // 16384 tiles total
  const int ti = tile >> 7;                   // / (NPTS/16 = 128)
  const int tj = tile & 127;
  const int r0 = ti * 16;
  const int c0 = tj * 16;

  const float* arow = X + (size_t)(r0 + l16) * NF + half * 2;
  const float* brow = X + (size_t)(c0 + l16) * NF + half * 2;

  v8f acc = {};
#pragma unroll 8
  for (int kb = 0; kb < NF; kb += 4) {
    const v2f a = *(const v2f*)(arow + kb);
    const v2f b = *(const v2f*)(brow + kb);
    acc = __builtin_amdgcn_wmma_f32_16x16x4_f32(
        /*neg_a=*/false, a, /*neg_b=*/false, b,
        /*c_mod=*/(short)0, acc, /*reuse_a=*/false, /*reuse_b=*/false);
  }

  const float inv2s = 0.5f / sigma2[0];       // 1 / (2*sigma^2)
  const float nj = norms[c0 + l16];
#pragma unroll
  for (int v = 0; v < 8; ++v) {
    const int row = r0 + v + half * 8;
    const float d2 = norms[row] + nj - 2.0f * acc[v];
    out[(size_t)row * NPTS + (c0 + l16)] = __expf(-d2 * inv2s);
  }
}

// ---------------------------------------------------------------------------
extern "C" void kernel_launch(void* const* d_in, const int* in_sizes, int n_in,
                              void* d_out, int out_size, void* d_ws,
                              size_t ws_size, hipStream_t stream) {
  (void)in_sizes; (void)n_in; (void)out_size; (void)ws_size;
  const float* X = (const float*)d_in[0];
  float* out = (float*)d_out;
  float* ws = (float*)d_ws;

  float* sigma2  = ws + WS_SIGMA;
  float* norms   = ws + WS_NORMS;
  float* colpart = ws + WS_COLS;

  norms_cols_kernel<<<K1_BLOCKS, 256, 0, stream>>>(X, norms, colpart);
  sigma_kernel<<<1, 256, 0, stream>>>(norms, colpart, sigma2);

  const int tiles = (NPTS / 16) * (NPTS / 16); // 16384
  gauss_wmma_kernel<<<tiles / 8, 256, 0, stream>>>(X, norms, sigma2, out);
}